// RelPositionMultiHeadedAttention_996432412941
// MI455X (gfx1250) — compile-verified
//
#include <hip/hip_runtime.h>
#include <math.h>

// ---------------------------------------------------------------------------
// RelPositionMultiHeadedAttention for MI455X (gfx1250), fp32 end-to-end using
// V_WMMA_F32_16X16X4_F32. Flash-style fused attention: scores never hit
// memory; rel_shift handled via a 31-row position window GEMM + LDS gather.
// ---------------------------------------------------------------------------

typedef float v2f __attribute__((ext_vector_type(2)));
typedef float v8f __attribute__((ext_vector_type(8)));

#define T_DIM 1024
#define B_DIM 4
#define F_DIM 512
#define H_DIM 8
#define DK_DIM 64
#define P_DIM 2047   // 2*T-1

__device__ __forceinline__ v8f wmma_f32(v2f a, v2f b, v8f c) {
  // D = A(16x4 f32) x B(4x16 f32) + C(16x16 f32)
  // args: (neg_a, A, neg_b, B, c_mod, C, reuse_a, reuse_b)
  return __builtin_amdgcn_wmma_f32_16x16x4_f32(false, a, false, b, (short)0, c,
                                               false, false);
}

// ---------------------------------------------------------------------------
// Projection: out[r,c] = sum_f in_row(r)[f] * W[c,f] + bias[c]
// q/k/v: input is (T,B,F) time-first; logical row r = b*T + t maps to input
// row t*B + b. Output stored batch-first (B*T, F) for attention tiling.
// One wave per 16x16 output tile. A and B operands are both contiguous
// float2 loads (W accessed transposed == row-major rows).
// ---------------------------------------------------------------------------
__global__ void proj_qkv_kernel(const float* __restrict__ in,   // (T,B,F)
                                const float* __restrict__ W,    // (F,F)
                                const float* __restrict__ bias, // (F)
                                float* __restrict__ out)        // (B*T,F)
{
  const int lane = threadIdx.x;
  const int lane16 = lane & 15;
  const int half = lane >> 4;
  const int r0 = blockIdx.x * 16;
  const int c0 = blockIdx.y * 16;

  const int gr = r0 + lane16;            // row in (b*T + t) space
  const int t = gr & (T_DIM - 1);
  const int b = gr >> 10;
  const float* aRow = in + (size_t)(t * B_DIM + b) * F_DIM;
  const float* bRow = W + (size_t)(c0 + lane16) * F_DIM;

  v8f acc = {};
#pragma unroll 8
  for (int k = 0; k < F_DIM; k += 4) {
    const int kk = k + 2 * half;
    v2f a = *(const v2f*)(aRow + kk);
    v2f bb = *(const v2f*)(bRow + kk);
    acc = wmma_f32(a, bb, acc);
  }
  const float bval = bias[c0 + lane16];
#pragma unroll
  for (int r = 0; r < 8; ++r) {
    const int M = r + 8 * half;
    out[(size_t)(r0 + M) * F_DIM + (c0 + lane16)] = acc[r] + bval;
  }
}

// Positional projection: p = pos_emb @ Wpos.T, rows 0..P-1 (no bias).
__global__ void proj_pos_kernel(const float* __restrict__ pe,   // (P,F)
                                const float* __restrict__ W,    // (F,F)
                                float* __restrict__ out)        // (P,F)
{
  const int lane = threadIdx.x;
  const int lane16 = lane & 15;
  const int half = lane >> 4;
  const int r0 = blockIdx.x * 16;
  const int c0 = blockIdx.y * 16;

  const int ar = (r0 + lane16 < P_DIM) ? (r0 + lane16) : (P_DIM - 1);
  const float* aRow = pe + (size_t)ar * F_DIM;
  const float* bRow = W + (size_t)(c0 + lane16) * F_DIM;

  v8f acc = {};
#pragma unroll 8
  for (int k = 0; k < F_DIM; k += 4) {
    const int kk = k + 2 * half;
    v2f a = *(const v2f*)(aRow + kk);
    v2f bb = *(const v2f*)(bRow + kk);
    acc = wmma_f32(a, bb, acc);
  }
#pragma unroll
  for (int r = 0; r < 8; ++r) {
    const int M = r + 8 * half;
    if (r0 + M < P_DIM)
      out[(size_t)(r0 + M) * F_DIM + (c0 + lane16)] = acc[r];
  }
}

// Output projection: attn (B*T,F) @ Wout.T + bout -> d_out (T,B,F)
__global__ void proj_out_kernel(const float* __restrict__ in,   // (B*T,F)
                                const float* __restrict__ W,    // (F,F)
                                const float* __restrict__ bias, // (F)
                                float* __restrict__ out)        // (T,B,F)
{
  const int lane = threadIdx.x;
  const int lane16 = lane & 15;
  const int half = lane >> 4;
  const int r0 = blockIdx.x * 16;
  const int c0 = blockIdx.y * 16;

  const float* aRow = in + (size_t)(r0 + lane16) * F_DIM;
  const float* bRow = W + (size_t)(c0 + lane16) * F_DIM;

  v8f acc = {};
#pragma unroll 8
  for (int k = 0; k < F_DIM; k += 4) {
    const int kk = k + 2 * half;
    v2f a = *(const v2f*)(aRow + kk);
    v2f bb = *(const v2f*)(bRow + kk);
    acc = wmma_f32(a, bb, acc);
  }
  const float bval = bias[c0 + lane16];
#pragma unroll
  for (int r = 0; r < 8; ++r) {
    const int row = r0 + r + 8 * half;           // b*T + t
    const int t = row & (T_DIM - 1);
    const int b = row >> 10;
    out[((size_t)t * B_DIM + b) * F_DIM + (c0 + lane16)] = acc[r] + bval;
  }
}

// ---------------------------------------------------------------------------
// Fused relative-position attention, flash style. One wave per
// (b, h, 16-row t-tile). Online softmax over 16-col s-tiles.
//   ac[t,s]  = q_u[t] . k[s]
//   bd[t,s]  = q_v[t] . p[T-1-t+s]   (rel_shift folded into the p index)
// For a tile: base = T-1-t0+s0; G[M][u] = q_v[t0+M] . p[base-15+u], u in [0,31)
// then bd[M][N] = G[M][15+N-M]  (gather through LDS).
// ---------------------------------------------------------------------------
__global__ void relattn_kernel(const float* __restrict__ qp,  // (B*T,F)
                               const float* __restrict__ kp,  // (B*T,F)
                               const float* __restrict__ vp,  // (B*T,F)
                               const float* __restrict__ pp,  // (P,F)
                               const float* __restrict__ pbu, // (H,DK)
                               const float* __restrict__ pbv, // (H,DK)
                               float* __restrict__ attn_out)  // (B*T,F)
{
  const int lane = threadIdx.x;
  const int lane16 = lane & 15;
  const int half = lane >> 4;
  const int t0 = blockIdx.x * 16;
  const int h = blockIdx.y;
  const int b = blockIdx.z;

  __shared__ float qu[16 * 66];   // q + pos_bias_u, padded stride
  __shared__ float qv[16 * 66];   // q + pos_bias_v
  __shared__ float G[16 * 34];    // 16x32 position-score strip
  __shared__ float Pt[16 * 18];   // softmaxed P tile staged as A operand

  for (int idx = lane; idx < 16 * 64; idx += 32) {
    const int i = idx >> 6, d = idx & 63;
    const float q = qp[(size_t)(b * T_DIM + t0 + i) * F_DIM + h * DK_DIM + d];
    qu[i * 66 + d] = q + pbu[h * DK_DIM + d];
    qv[i * 66 + d] = q + pbv[h * DK_DIM + d];
  }
  __syncthreads();

  float mstate[8], lstate[8];
  v8f acc[4];
#pragma unroll
  for (int r = 0; r < 8; ++r) { mstate[r] = -INFINITY; lstate[r] = 0.0f; }
#pragma unroll
  for (int nt = 0; nt < 4; ++nt) acc[nt] = (v8f){};

  for (int s0 = 0; s0 < T_DIM; s0 += 16) {
    // ---- content scores: q_u_tile @ k_tile^T (K = 64) ----
    const float* kRow = kp + (size_t)(b * T_DIM + s0 + lane16) * F_DIM + h * DK_DIM;
    v8f Sc = {};
#pragma unroll
    for (int k = 0; k < DK_DIM; k += 4) {
      const int kk = k + 2 * half;
      v2f a = *(const v2f*)(&qu[lane16 * 66 + kk]);
      v2f bb = *(const v2f*)(kRow + kk);
      Sc = wmma_f32(a, bb, Sc);
    }

    // ---- position-score strip over the 31-row p window ----
    const int base = (T_DIM - 1) - t0 + s0;       // >= 15 always
    const int pidxL = base - 15 + lane16;          // in [0, P)
    const int pidxR = base + 1 + lane16;           // pad col may hit P
    const bool rValid = (pidxR < P_DIM);
    const float* pRowL = pp + (size_t)pidxL * F_DIM + h * DK_DIM;
    const float* pRowR = pp + (size_t)(rValid ? pidxR : (P_DIM - 1)) * F_DIM + h * DK_DIM;

    v8f Gl = {}, Gr = {};
#pragma unroll
    for (int k = 0; k < DK_DIM; k += 4) {
      const int kk = k + 2 * half;
      v2f a = *(const v2f*)(&qv[lane16 * 66 + kk]);
      v2f bl = *(const v2f*)(pRowL + kk);
      v2f br = *(const v2f*)(pRowR + kk);
      if (!rValid) { br.x = 0.0f; br.y = 0.0f; }
      Gl = wmma_f32(a, bl, Gl);
      Gr = wmma_f32(a, br, Gr);
    }
#pragma unroll
    for (int r = 0; r < 8; ++r) {
      const int M = r + 8 * half;
      G[M * 34 + lane16] = Gl[r];
      G[M * 34 + 16 + lane16] = Gr[r];
    }
    __syncthreads();

    // ---- combine + rel-shift gather + scale ----
    float S[8];
#pragma unroll
    for (int r = 0; r < 8; ++r) {
      const int M = r + 8 * half;
      const int c = 15 + lane16 - M;              // in [0, 30]
      S[r] = (Sc[r] + G[M * 34 + c]) * 0.125f;    // 1/sqrt(64)
    }
    __syncthreads();

    // ---- online softmax (rows live across 16 lanes of a half) ----
    float Pexp[8];
#pragma unroll
    for (int r = 0; r < 8; ++r) {
      float v = S[r];
      for (int off = 8; off >= 1; off >>= 1) v = fmaxf(v, __shfl_xor(v, off, 16));
      const float mnew = fmaxf(mstate[r], v);
      const float alpha = __expf(mstate[r] - mnew);
      const float pe = __expf(S[r] - mnew);
      float sum = pe;
      for (int off = 8; off >= 1; off >>= 1) sum += __shfl_xor(sum, off, 16);
      lstate[r] = lstate[r] * alpha + sum;
      mstate[r] = mnew;
      Pexp[r] = pe;
#pragma unroll
      for (int nt = 0; nt < 4; ++nt) acc[nt][r] *= alpha;
    }

    // ---- stage P tile as A operand ----
#pragma unroll
    for (int r = 0; r < 8; ++r) {
      const int M = r + 8 * half;
      Pt[M * 18 + lane16] = Pexp[r];
    }
    __syncthreads();

    // ---- acc += P_tile @ V_tile (K = 16, 4 d-subtiles) ----
#pragma unroll
    for (int nt = 0; nt < 4; ++nt) {
      const float* vCol =
          vp + (size_t)(b * T_DIM + s0) * F_DIM + h * DK_DIM + nt * 16 + lane16;
      v8f a4 = acc[nt];
#pragma unroll
      for (int k = 0; k < 16; k += 4) {
        const int kk = k + 2 * half;
        v2f a = *(const v2f*)(&Pt[lane16 * 18 + kk]);
        v2f bb;
        bb.x = vCol[(size_t)kk * F_DIM];
        bb.y = vCol[(size_t)(kk + 1) * F_DIM];
        a4 = wmma_f32(a, bb, a4);
      }
      acc[nt] = a4;
    }
    __syncthreads();  // G/Pt reused next iteration
  }

  // ---- normalize and write (B*T, F) with f = h*DK + d ----
#pragma unroll
  for (int r = 0; r < 8; ++r) {
    const float inv = 1.0f / lstate[r];
    const int M = r + 8 * half;
#pragma unroll
    for (int nt = 0; nt < 4; ++nt) {
      attn_out[(size_t)(b * T_DIM + t0 + M) * F_DIM + h * DK_DIM + nt * 16 +
               lane16] = acc[nt][r] * inv;
    }
  }
}

// ---------------------------------------------------------------------------
// Launch. Workspace layout (floats):
//   qp: 4096*512 | kp: 4096*512 | vp: 4096*512 | attn: 4096*512 | pp: 2047*512
// total ~37.8 MB.
// ---------------------------------------------------------------------------
extern "C" void kernel_launch(void* const* d_in, const int* in_sizes, int n_in,
                              void* d_out, int out_size, void* d_ws,
                              size_t ws_size, hipStream_t stream) {
  const float* query   = (const float*)d_in[0];
  const float* key     = (const float*)d_in[1];
  const float* value   = (const float*)d_in[2];
  const float* pos_emb = (const float*)d_in[3];
  const float* Wq      = (const float*)d_in[4];
  const float* bq      = (const float*)d_in[5];
  const float* Wk      = (const float*)d_in[6];
  const float* bk      = (const float*)d_in[7];
  const float* Wv      = (const float*)d_in[8];
  const float* bv      = (const float*)d_in[9];
  const float* Wpos    = (const float*)d_in[10];
  const float* Wout    = (const float*)d_in[11];
  const float* bout    = (const float*)d_in[12];
  const float* pbu     = (const float*)d_in[13];
  const float* pbv     = (const float*)d_in[14];

  float* ws = (float*)d_ws;
  const size_t nBT = (size_t)B_DIM * T_DIM * F_DIM;  // 2,097,152
  float* qp   = ws;
  float* kp   = qp + nBT;
  float* vp   = kp + nBT;
  float* attn = vp + nBT;
  float* pp   = attn + nBT;   // 2047*512

  const dim3 blk(32);
  const dim3 gProj(B_DIM * T_DIM / 16, F_DIM / 16);   // (256, 32)
  const dim3 gPos((P_DIM + 15) / 16, F_DIM / 16);     // (128, 32)
  const dim3 gAttn(T_DIM / 16, H_DIM, B_DIM);         // (64, 8, 4)

  proj_qkv_kernel<<<gProj, blk, 0, stream>>>(query, Wq, bq, qp);
  proj_qkv_kernel<<<gProj, blk, 0, stream>>>(key, Wk, bk, kp);
  proj_qkv_kernel<<<gProj, blk, 0, stream>>>(value, Wv, bv, vp);
  proj_pos_kernel<<<gPos, blk, 0, stream>>>(pos_emb, Wpos, pp);
  relattn_kernel<<<gAttn, blk, 0, stream>>>(qp, kp, vp, pp, pbu, pbv, attn);
  proj_out_kernel<<<gProj, blk, 0, stream>>>(attn, Wout, bout, (float*)d_out);
}